// SimpleGINLayer_44744969290327
// MI455X (gfx1250) — compile-verified
//
#include <hip/hip_runtime.h>

typedef __attribute__((ext_vector_type(2))) float v2f;
typedef __attribute__((ext_vector_type(8))) float v8f;

#define D_FEAT 128

// ---------------------------------------------------------------------------
// Kernel 1: out[v, :] = (1+eps) * norm[v] * h[v, :]   via WMMA diag-scale.
// Block = 256 threads = 8 waves. Block nb owns node rows [nb*16, nb*16+16).
// Wave w owns the 16-column tile starting at col w*16 (D_FEAT = 8 tiles).
// Per tile: D(16x16) = sum_{c=0..3} A_c(16x4) x B_c(4x16), accumulator-chained.
//   A_c[m,k] = s_m  if m == 4c+k else 0      (s_m = (1+eps)*norm[row0+m])
//   B_c[k,n] = h[row0 + 4c + k, col0 + n]
// => D[m,n] = s_m * h[row0+m, col0+n].
// A lane->M map and B/D lane->N map follow the ISA tables; the k-slot
// convention is shared between A and B so any slot permutation cancels.
// ---------------------------------------------------------------------------
__global__ __launch_bounds__(256) void gin_init_wmma(
    const float* __restrict__ h, const float* __restrict__ norm,
    const float* __restrict__ eps, float* __restrict__ out) {
  const int nb   = blockIdx.x;
  const int lane = threadIdx.x & 31;
  const int wave = threadIdx.x >> 5;
  const int row0 = nb * 16;
  const int col0 = wave * 16;
  const int m    = lane & 15;   // A: row M of tile.  B/D: column N of tile.
  const int kh   = lane >> 4;   // lane half selects the k-pair slot
  const int k0   = kh * 2;
  const float sm = (1.0f + eps[0]) * norm[row0 + m];

  v8f acc = {};
#pragma unroll
  for (int c = 0; c < 4; ++c) {
    v2f a, b;
    a[0] = (m == 4 * c + k0)     ? sm : 0.0f;
    a[1] = (m == 4 * c + k0 + 1) ? sm : 0.0f;
    b[0] = h[(size_t)(row0 + 4 * c + k0)     * D_FEAT + col0 + m];
    b[1] = h[(size_t)(row0 + 4 * c + k0 + 1) * D_FEAT + col0 + m];
    // 8 args: (neg_a, A, neg_b, B, c_mod, C, reuse_a, reuse_b)
    acc = __builtin_amdgcn_wmma_f32_16x16x4_f32(false, a, false, b,
                                                (short)0, acc, false, false);
  }
  // D layout: VGPR r -> M = r (lanes 0-15) / r+8 (lanes 16-31), N = lane&15.
#pragma unroll
  for (int r = 0; r < 8; ++r) {
    out[(size_t)(row0 + r + 8 * kh) * D_FEAT + col0 + m] = acc[r];
  }
}

// Scalar tail for node counts not divisible by 16 (unused for N=50000).
__global__ void gin_init_tail(const float* __restrict__ h,
                              const float* __restrict__ norm,
                              const float* __restrict__ eps,
                              float* __restrict__ out,
                              int row_start, int n_nodes) {
  int i = blockIdx.x * blockDim.x + threadIdx.x;
  int total = (n_nodes - row_start) * D_FEAT;
  if (i >= total) return;
  int node = row_start + i / D_FEAT;
  int col  = i % D_FEAT;
  out[(size_t)node * D_FEAT + col] =
      (1.0f + eps[0]) * norm[node] * h[(size_t)node * D_FEAT + col];
}

// ---------------------------------------------------------------------------
// Kernel 2: scatter-add.  One wave32 per edge; lane l handles floats
// [4l, 4l+4) of the 128-float row.  h and the accumulator both live in the
// 192 MB L2, so the random gather + atomics stay on-chip.
// ---------------------------------------------------------------------------
__global__ __launch_bounds__(256) void gin_edge_scatter(
    const float* __restrict__ h, const float* __restrict__ norm,
    const int* __restrict__ src, const int* __restrict__ dst,
    float* __restrict__ out, int n_edges) {
  const int lane = threadIdx.x & 31;
  const int e = blockIdx.x * (blockDim.x >> 5) + (threadIdx.x >> 5);
  if (e >= n_edges) return;
  const int s = src[e];
  const int d = dst[e];
  const float ns = norm[s];
  const float4 v = ((const float4*)(h + (size_t)s * D_FEAT))[lane];
  float* orow = out + (size_t)d * D_FEAT + lane * 4;
  atomicAdd(orow + 0, v.x * ns);   // global_atomic_add_f32, no return
  atomicAdd(orow + 1, v.y * ns);
  atomicAdd(orow + 2, v.z * ns);
  atomicAdd(orow + 3, v.w * ns);
}

// ---------------------------------------------------------------------------
// Kernel 3: out[v, :] *= norm[v]   (post-normalize), float4 vectorized.
// ---------------------------------------------------------------------------
__global__ __launch_bounds__(256) void gin_post_scale(
    const float* __restrict__ norm, float* __restrict__ out, int n_nodes) {
  int t = blockIdx.x * blockDim.x + threadIdx.x;   // one float4 per thread
  int node = t >> 5;                               // 32 float4 per row
  if (node >= n_nodes) return;
  const float nv = norm[node];
  float4* p = (float4*)out + t;
  float4 v = *p;
  v.x *= nv; v.y *= nv; v.z *= nv; v.w *= nv;
  *p = v;
}

extern "C" void kernel_launch(void* const* d_in, const int* in_sizes, int n_in,
                              void* d_out, int out_size, void* d_ws, size_t ws_size,
                              hipStream_t stream) {
  const float* h    = (const float*)d_in[0];
  const float* norm = (const float*)d_in[1];
  const float* eps  = (const float*)d_in[2];
  const int*   src  = (const int*)d_in[3];
  const int*   dst  = (const int*)d_in[4];
  float* out = (float*)d_out;

  const int n_nodes = in_sizes[1];   // norm: one entry per node
  const int n_edges = in_sizes[3];

  // Phase 1: init accumulator out = (1+eps) * h * norm
  const int full = n_nodes & ~15;
  if (full > 0) {
    gin_init_wmma<<<full / 16, 256, 0, stream>>>(h, norm, eps, out);
  }
  if (n_nodes - full > 0) {
    int tot = (n_nodes - full) * D_FEAT;
    gin_init_tail<<<(tot + 255) / 256, 256, 0, stream>>>(h, norm, eps, out,
                                                         full, n_nodes);
  }

  // Phase 2: scatter-add neighbor features (one wave per edge, 8 waves/block)
  const int eb = (n_edges + 7) / 8;
  gin_edge_scatter<<<eb, 256, 0, stream>>>(h, norm, src, dst, out, n_edges);

  // Phase 3: post-normalize
  const int q = n_nodes * (D_FEAT / 4);
  gin_post_scale<<<(q + 255) / 256, 256, 0, stream>>>(norm, out, n_nodes);
}